// PreintegrationModule_59871844106883
// MI455X (gfx1250) — compile-verified
//
#include <hip/hip_runtime.h>
#include <hip/hip_bf16.h>
#include <stdint.h>

// ---- problem constants (from reference) -------------------------------------
#define IMU_B 4096
#define IMU_S 2048
#define IMU_DT 0.005f
#define IMU_HALF_DT2 (IMU_DT * IMU_DT * 0.5f)

// ---- decomposition ----------------------------------------------------------
#define NCHUNK 16                         // chunks per batch (parallel over S)
#define CHUNK_LEN (IMU_S / NCHUNK)        // 128 steps per chunk
#define TILE_STEPS 8                      // steps staged per TDM tile
#define NTILES (CHUNK_LEN / TILE_STEPS)   // 16 tiles per chunk
#define WAVES_PER_BLOCK 4
#define BLOCK1 (WAVES_PER_BLOCK * 32)     // 128 threads (wave32)
#define ROW_FLOATS (TILE_STEPS * 6)       // 48 floats = 192 B per lane-row
#define BUF_FLOATS (32 * ROW_FLOATS)      // 1536 floats = 6144 B per buffer
#define BUF_BYTES (BUF_FLOATS * 4)
#define CHUNK_BYTES (CHUNK_LEN * 6 * 4)   // 3072 B per chunk (row stride)

typedef unsigned int u32x4 __attribute__((ext_vector_type(4)));
typedef int i32x8 __attribute__((ext_vector_type(8)));
typedef int i32x4 __attribute__((ext_vector_type(4)));

// ---- one integration step (Rodrigues + compose + integrate) -----------------
__device__ __forceinline__ void imu_step(float wx, float wy, float wz,
                                         float ax, float ay, float az,
                                         float dr[9], float dv[3], float dp[3]) {
    const float dt = IMU_DT;
    const float h2 = IMU_HALF_DT2;
    float n2 = wx * wx + wy * wy + wz * wz;
    float n = sqrtf(n2);
    float th = n * dt;
    float s = __sinf(th);
    float c = __cosf(th);
    float ns = fmaxf(n, 1e-12f);
    float inv = 1.0f / ns;
    float Af = s * inv;                 // sin(th)/|w|
    float Bf = (1.0f - c) * inv * inv;  // (1-cos(th))/|w|^2
    // R = c*I + Bf*w w^T + Af*skew(w)
    float R00 = c + Bf * wx * wx;
    float R01 = Bf * wx * wy - Af * wz;
    float R02 = Bf * wx * wz + Af * wy;
    float R10 = Bf * wx * wy + Af * wz;
    float R11 = c + Bf * wy * wy;
    float R12 = Bf * wy * wz - Af * wx;
    float R20 = Bf * wx * wz - Af * wy;
    float R21 = Bf * wy * wz + Af * wx;
    float R22 = c + Bf * wz * wz;
    // dr = dr @ R
    float t0, t1, t2;
    t0 = dr[0] * R00 + dr[1] * R10 + dr[2] * R20;
    t1 = dr[0] * R01 + dr[1] * R11 + dr[2] * R21;
    t2 = dr[0] * R02 + dr[1] * R12 + dr[2] * R22;
    dr[0] = t0; dr[1] = t1; dr[2] = t2;
    t0 = dr[3] * R00 + dr[4] * R10 + dr[5] * R20;
    t1 = dr[3] * R01 + dr[4] * R11 + dr[5] * R21;
    t2 = dr[3] * R02 + dr[4] * R12 + dr[5] * R22;
    dr[3] = t0; dr[4] = t1; dr[5] = t2;
    t0 = dr[6] * R00 + dr[7] * R10 + dr[8] * R20;
    t1 = dr[6] * R01 + dr[7] * R11 + dr[8] * R21;
    t2 = dr[6] * R02 + dr[7] * R12 + dr[8] * R22;
    dr[6] = t0; dr[7] = t1; dr[8] = t2;
    // m = dr @ a ; dv += m*dt ; dp += dv*dt + m*half_dt2
    float m0 = dr[0] * ax + dr[1] * ay + dr[2] * az;
    float m1 = dr[3] * ax + dr[4] * ay + dr[5] * az;
    float m2 = dr[6] * ax + dr[7] * ay + dr[8] * az;
    dv[0] += m0 * dt; dv[1] += m1 * dt; dv[2] += m2 * dt;
    dp[0] += dv[0] * dt + m0 * h2;
    dp[1] += dv[1] * dt + m1 * h2;
    dp[2] += dv[2] * dt + m2 * h2;
}

// ---- rotation-matrix -> quaternion + pack output ----------------------------
__device__ __forceinline__ void finalize_out(const float A[9], const float pos[3],
                                             float* o) {
    float tr = A[0] + A[4] + A[8];
    float ca = (tr - 1.0f) * 0.5f;
    ca = fminf(fmaxf(ca, -1.0f + 1e-7f), 1.0f - 1e-7f);
    float ang = acosf(ca);
    float den = fmaxf(2.0f * __sinf(ang), 1e-12f);
    float inv = 1.0f / den;
    float axx = (A[7] - A[5]) * inv;
    float axy = (A[2] - A[6]) * inv;
    float axz = (A[3] - A[1]) * inv;
    float half = 0.5f * ang;
    float sh = __sinf(half), ch = __cosf(half);
    o[0] = pos[0]; o[1] = pos[1]; o[2] = pos[2];
    o[3] = ch; o[4] = axx * sh; o[5] = axy * sh; o[6] = axz * sh;
}

// ---- TDM issue: 2D tile, 32 rows x 192 B, row stride 3072 B -----------------
__device__ __forceinline__ void tdm_issue(uint64_t gaddr, unsigned ldsOff) {
    // D# group 0: count=1 | lds_addr | global_addr[56:0] | type=2
    u32x4 g0 = { 1u, ldsOff, (unsigned)(gaddr & 0xFFFFFFFFu),
                 (unsigned)((gaddr >> 32) & 0x1FFFFFFu) | (2u << 30) };
    // D# group 1: data_size=4B; tensor_dim0/1 = 2^30 (no OOB); tile 48x32;
    // tensor_dim0_stride = 768 elements (3072 B)
    i32x8 g1 = { (int)(2u << 16),                 // data_size=2 (4B), mask=0
                 0,                               // atomic_barrier_addr=0, dim0 lo16=0
                 0x4000,                          // dim0 hi16=0x4000, dim1 lo16=0
                 (int)(0x4000u | (48u << 16)),    // dim1 hi16, tile_dim0=48
                 32,                              // tile_dim1=32, tile_dim2=0
                 (int)(CHUNK_LEN * 6),            // tensor_dim0_stride=768
                 0, 0 };
    i32x4 g2 = { 0, 0, 0, 0 };
    i32x4 g3 = { 0, 0, 0, 0 };
#if defined(__clang_major__) && (__clang_major__ >= 23)
    // amdgpu-toolchain (clang-23 / therock-10.0): 6-arg form
    i32x8 gpad = { 0, 0, 0, 0, 0, 0, 0, 0 };
    __builtin_amdgcn_tensor_load_to_lds(g0, g1, g2, g3, gpad, 0);
#else
    // ROCm 7.2 (clang-22): 5-arg form
    __builtin_amdgcn_tensor_load_to_lds(g0, g1, g2, g3, 0);
#endif
}

// ---- kernel 1: per-(batch,chunk) partial scan, TDM double-buffered ----------
__global__ __launch_bounds__(BLOCK1) void preint_partial_kernel(
        const float* __restrict__ in, float* __restrict__ ws) {
    __shared__ __align__(16) float smem[WAVES_PER_BLOCK * 2 * BUF_FLOATS]; // 48 KB
    const int lane = threadIdx.x & 31;
    const int wid = threadIdx.x >> 5;
    const int gw = blockIdx.x * WAVES_PER_BLOCK + wid;  // global wave id
    const int g = gw * 32 + lane;                       // batch*NCHUNK + chunk

    unsigned ldsBase;
    {
        typedef __attribute__((address_space(3))) char lds_char_t;
        ldsBase = (unsigned)(unsigned long long)(lds_char_t*)&smem[0];
    }
    const uint64_t srcBase =
        (uint64_t)(uintptr_t)in + (uint64_t)gw * (32ull * CHUNK_BYTES);

    float dr[9] = {1.f, 0.f, 0.f, 0.f, 1.f, 0.f, 0.f, 0.f, 1.f};
    float dv[3] = {0.f, 0.f, 0.f};
    float dp[3] = {0.f, 0.f, 0.f};

    // prologue: stage tile 0
    tdm_issue(srcBase, ldsBase + (unsigned)(wid * 2 + 0) * BUF_BYTES);
    for (int t = 0; t < NTILES; ++t) {
        const int buf = t & 1;
        if (t + 1 < NTILES) {
            tdm_issue(srcBase + (uint64_t)(t + 1) * (TILE_STEPS * 24u),
                      ldsBase + (unsigned)(wid * 2 + ((t + 1) & 1)) * BUF_BYTES);
            __builtin_amdgcn_s_wait_tensorcnt(1);  // oldest (tile t) complete
        } else {
            __builtin_amdgcn_s_wait_tensorcnt(0);
        }
        asm volatile("" ::: "memory");
        const float* row = &smem[(wid * 2 + buf) * BUF_FLOATS + lane * ROW_FLOATS];
#pragma unroll
        for (int s = 0; s < TILE_STEPS; ++s) {
            const float* rp = row + s * 6;
            float2 p0 = *(const float2*)(rp);
            float2 p1 = *(const float2*)(rp + 2);
            float2 p2 = *(const float2*)(rp + 4);
            imu_step(p0.x, p0.y, p1.x, p1.y, p2.x, p2.y, dr, dv, dp);
        }
        asm volatile("" ::: "memory");
    }

    float* e = ws + (size_t)g * 16;
    *(float4*)(e)      = make_float4(dr[0], dr[1], dr[2], dr[3]);
    *(float4*)(e + 4)  = make_float4(dr[4], dr[5], dr[6], dr[7]);
    *(float4*)(e + 8)  = make_float4(dr[8], dv[0], dv[1], dv[2]);
    *(float4*)(e + 12) = make_float4(dp[0], dp[1], dp[2], 0.0f);
}

// ---- kernel 2: fold NCHUNK partials per batch (associative composition) -----
__global__ __launch_bounds__(128) void preint_reduce_kernel(
        const float* __restrict__ ws, float* __restrict__ out) {
    const int b = blockIdx.x * blockDim.x + threadIdx.x;
    if (b >= IMU_B) return;
    float A[9] = {1.f, 0.f, 0.f, 0.f, 1.f, 0.f, 0.f, 0.f, 1.f};
    float vel[3] = {0.f, 0.f, 0.f};
    float pos[3] = {0.f, 0.f, 0.f};
    const float Ldt = (float)CHUNK_LEN * IMU_DT;
    for (int c = 0; c < NCHUNK; ++c) {
        const float* e = ws + ((size_t)b * NCHUNK + c) * 16;
        float4 q0 = *(const float4*)(e);
        float4 q1 = *(const float4*)(e + 4);
        float4 q2 = *(const float4*)(e + 8);
        float4 q3 = *(const float4*)(e + 12);
        float R[9] = {q0.x, q0.y, q0.z, q0.w, q1.x, q1.y, q1.z, q1.w, q2.x};
        float b2[3] = {q2.y, q2.z, q2.w};
        float d2[3] = {q3.x, q3.y, q3.z};
        // pos += vel*L*dt + A@d2 ; vel += A@b2 ; A = A@R
        float Ad0 = A[0] * d2[0] + A[1] * d2[1] + A[2] * d2[2];
        float Ad1 = A[3] * d2[0] + A[4] * d2[1] + A[5] * d2[2];
        float Ad2 = A[6] * d2[0] + A[7] * d2[1] + A[8] * d2[2];
        float Ab0 = A[0] * b2[0] + A[1] * b2[1] + A[2] * b2[2];
        float Ab1 = A[3] * b2[0] + A[4] * b2[1] + A[5] * b2[2];
        float Ab2 = A[6] * b2[0] + A[7] * b2[1] + A[8] * b2[2];
        pos[0] += vel[0] * Ldt + Ad0;
        pos[1] += vel[1] * Ldt + Ad1;
        pos[2] += vel[2] * Ldt + Ad2;
        vel[0] += Ab0; vel[1] += Ab1; vel[2] += Ab2;
        float nA[9];
        nA[0] = A[0] * R[0] + A[1] * R[3] + A[2] * R[6];
        nA[1] = A[0] * R[1] + A[1] * R[4] + A[2] * R[7];
        nA[2] = A[0] * R[2] + A[1] * R[5] + A[2] * R[8];
        nA[3] = A[3] * R[0] + A[4] * R[3] + A[5] * R[6];
        nA[4] = A[3] * R[1] + A[4] * R[4] + A[5] * R[7];
        nA[5] = A[3] * R[2] + A[4] * R[5] + A[5] * R[8];
        nA[6] = A[6] * R[0] + A[7] * R[3] + A[8] * R[6];
        nA[7] = A[6] * R[1] + A[7] * R[4] + A[8] * R[7];
        nA[8] = A[6] * R[2] + A[7] * R[5] + A[8] * R[8];
#pragma unroll
        for (int k = 0; k < 9; ++k) A[k] = nA[k];
    }
    finalize_out(A, pos, out + (size_t)b * 7);
}

// ---- fallback: one thread per batch, full serial scan (no workspace) --------
__global__ __launch_bounds__(256) void preint_mono_kernel(
        const float* __restrict__ in, float* __restrict__ out) {
    const int b = blockIdx.x * blockDim.x + threadIdx.x;
    if (b >= IMU_B) return;
    const float* p = in + (size_t)b * IMU_S * 6;
    float dr[9] = {1.f, 0.f, 0.f, 0.f, 1.f, 0.f, 0.f, 0.f, 1.f};
    float dv[3] = {0.f, 0.f, 0.f};
    float dp[3] = {0.f, 0.f, 0.f};
    for (int s = 0; s < IMU_S; ++s) {
        const float* rp = p + s * 6;
        float2 p0 = *(const float2*)(rp);
        float2 p1 = *(const float2*)(rp + 2);
        float2 p2 = *(const float2*)(rp + 4);
        imu_step(p0.x, p0.y, p1.x, p1.y, p2.x, p2.y, dr, dv, dp);
    }
    finalize_out(dr, dp, out + (size_t)b * 7);
}

extern "C" void kernel_launch(void* const* d_in, const int* in_sizes, int n_in,
                              void* d_out, int out_size, void* d_ws, size_t ws_size,
                              hipStream_t stream) {
    (void)in_sizes; (void)n_in; (void)out_size;
    const float* in = (const float*)d_in[0];
    float* out = (float*)d_out;
    const size_t need = (size_t)IMU_B * NCHUNK * 16 * sizeof(float);  // 4 MB
    if (d_ws != nullptr && ws_size >= need) {
        float* ws = (float*)d_ws;
        preint_partial_kernel<<<(IMU_B * NCHUNK) / BLOCK1, BLOCK1, 0, stream>>>(in, ws);
        preint_reduce_kernel<<<IMU_B / 128, 128, 0, stream>>>(ws, out);
    } else {
        preint_mono_kernel<<<IMU_B / 256, 256, 0, stream>>>(in, out);
    }
}